// hgnn_6932077216397
// MI455X (gfx1250) — compile-verified
//
#include <hip/hip_runtime.h>
#include <hip/hip_bf16.h>

typedef float v2f __attribute__((ext_vector_type(2)));
typedef float v8f __attribute__((ext_vector_type(8)));

#define TPB 256

// ---------------- utility kernels ----------------

__global__ void hgnn_zero_f4(float4* __restrict__ p, unsigned n4) {
    unsigned i = blockIdx.x * blockDim.x + threadIdx.x;
    if (i < n4) p[i] = make_float4(0.f, 0.f, 0.f, 0.f);
}

__global__ void hgnn_deg_pair(const int* __restrict__ src, const int* __restrict__ dst,
                              float* __restrict__ dsrc, float* __restrict__ ddst, int E) {
    int e = blockIdx.x * blockDim.x + threadIdx.x;
    if (e < E) {
        atomicAdd(&dsrc[src[e]], 1.0f);
        atomicAdd(&ddst[dst[e]], 1.0f);
    }
}

__global__ void hgnn_deg_dst(const int* __restrict__ dst, float* __restrict__ ddst, int E) {
    int e = blockIdx.x * blockDim.x + threadIdx.x;
    if (e < E) atomicAdd(&ddst[dst[e]], 1.0f);
}

// in-place: degree counts -> rsqrt / reciprocal scale factors
__global__ void hgnn_finalize_scales(float* __restrict__ rs_out_pre, float* __restrict__ rs_in_pre,
                                     float* __restrict__ rs_out_nxt, float* __restrict__ rs_in_nxt,
                                     float* __restrict__ inv_cnt, int n) {
    int i = blockIdx.x * blockDim.x + threadIdx.x;
    if (i >= n) return;
    rs_out_pre[i] = rsqrtf(fmaxf(rs_out_pre[i], 1.0f));
    rs_in_pre[i]  = rsqrtf(fmaxf(rs_in_pre[i],  1.0f));
    rs_out_nxt[i] = rsqrtf(fmaxf(rs_out_nxt[i], 1.0f));
    rs_in_nxt[i]  = rsqrtf(fmaxf(rs_in_nxt[i],  1.0f));
    inv_cnt[i]    = 1.0f / fmaxf(inv_cnt[i], 1.0f);
}

// generic dense embedding: out[n,16] = feat[n,kin] @ W[kin,16] + b[16]
__global__ void hgnn_embed(const float* __restrict__ feat, const float* __restrict__ W,
                           const float* __restrict__ b, float* __restrict__ out,
                           int n, int kin) {
    int t = blockIdx.x * blockDim.x + threadIdx.x;
    if (t >= n * 16) return;
    int node = t >> 4, f = t & 15;
    float acc = b[f];
    const float* frow = feat + (size_t)node * kin;
    for (int k = 0; k < kin; ++k) acc = fmaf(frow[k], W[k * 16 + f], acc);
    out[t] = acc;
}

// scatter-add: agg[dst[e], f] += h_src[src[e], f] * (scale ? scale[src[e]] : 1)
// one thread per (edge, feature): half-wave-coalesced index, gather, and atomic streams
__global__ void hgnn_scatter(const float* __restrict__ hsrc, const int* __restrict__ src,
                             const int* __restrict__ dst, const float* __restrict__ scale,
                             float* __restrict__ agg, unsigned nthreads) {
    unsigned t = blockIdx.x * blockDim.x + threadIdx.x;
    if (t >= nthreads) return;
    unsigned e = t >> 4, f = t & 15;
    int s = src[e], d = dst[e];
    float v = hsrc[(size_t)s * 16 + f];
    if (scale) v *= scale[s];
    atomicAdd(&agg[(size_t)d * 16 + f], v);
}

// ---------------- WMMA node-update kernel ----------------
// Per wave: a 16-node tile. D[16x16] = A_pre@W_pre + A_nxt@W_nxt + hj@W_self + A_sage@W_neigh,
// accumulated through 16 chained V_WMMA_F32_16X16X4_F32 into one C fragment, + summed bias.

__device__ __forceinline__ v2f hgnn_bfrag(const float* __restrict__ W, int k0, int koff, int col) {
    v2f b;
    b.x = W[(k0 + koff) * 16 + col];
    b.y = W[(k0 + koff + 1) * 16 + col];
    return b;
}

__device__ __forceinline__ v2f hgnn_afrag(const float* __restrict__ x, int row, int k0, int koff, float s) {
    const float2 t = *(const float2*)(x + (size_t)row * 16 + k0 + koff);
    v2f a;
    a.x = t.x * s;
    a.y = t.y * s;
    return a;
}

__global__ void __launch_bounds__(TPB)
hgnn_update_wmma(float* __restrict__ hj,
                 const float* __restrict__ agg_pre, const float* __restrict__ agg_nxt,
                 const float* __restrict__ agg_sage,
                 const float* __restrict__ rs_in_pre, const float* __restrict__ rs_in_nxt,
                 const float* __restrict__ inv_cnt,
                 const float* __restrict__ W_pre, const float* __restrict__ W_nxt,
                 const float* __restrict__ W_self, const float* __restrict__ W_neigh,
                 const float* __restrict__ b_pre, const float* __restrict__ b_nxt,
                 const float* __restrict__ b_sage, int n_nodes) {
    int wave = (int)((blockIdx.x * blockDim.x + threadIdx.x) >> 5);
    int lane = (int)(threadIdx.x & 31);
    int ntiles = (n_nodes + 15) >> 4;
    if (wave >= ntiles) return;   // whole-wave uniform exit; EXEC stays all-1s below

    int base = wave * 16;
    int mrow = base + (lane & 15);
    if (mrow >= n_nodes) mrow = n_nodes - 1;  // clamp loads, guard stores later
    int koff = (lane >= 16) ? 2 : 0;          // A/B K sub-offset per half-wave
    int col  = lane & 15;                     // B column / D column

    float s_pre = rs_in_pre[mrow];
    float s_nxt = rs_in_nxt[mrow];
    float s_sag = inv_cnt[mrow];

    v8f c = {0.f, 0.f, 0.f, 0.f, 0.f, 0.f, 0.f, 0.f};

#pragma unroll
    for (int kc = 0; kc < 4; ++kc) {
        const int k0 = kc * 4;
        v2f a, b;

        a = hgnn_afrag(agg_pre, mrow, k0, koff, s_pre);
        b = hgnn_bfrag(W_pre, k0, koff, col);
        c = __builtin_amdgcn_wmma_f32_16x16x4_f32(false, a, false, b, (short)0, c, false, false);

        a = hgnn_afrag(agg_nxt, mrow, k0, koff, s_nxt);
        b = hgnn_bfrag(W_nxt, k0, koff, col);
        c = __builtin_amdgcn_wmma_f32_16x16x4_f32(false, a, false, b, (short)0, c, false, false);

        a = hgnn_afrag(hj, mrow, k0, koff, 1.0f);
        b = hgnn_bfrag(W_self, k0, koff, col);
        c = __builtin_amdgcn_wmma_f32_16x16x4_f32(false, a, false, b, (short)0, c, false, false);

        a = hgnn_afrag(agg_sage, mrow, k0, koff, s_sag);
        b = hgnn_bfrag(W_neigh, k0, koff, col);
        c = __builtin_amdgcn_wmma_f32_16x16x4_f32(false, a, false, b, (short)0, c, false, false);
    }

    float bias = b_pre[col] + b_nxt[col] + b_sage[col];
    int moff = (lane >= 16) ? 8 : 0;   // C/D: M = vgpr + 8*(lane>=16), N = lane&15
#pragma unroll
    for (int g = 0; g < 8; ++g) {
        int node = base + g + moff;
        if (node < n_nodes) hj[(size_t)node * 16 + col] = c[g] + bias;
    }
}

// ---------------- readout ----------------

__global__ void hgnn_score(const float* __restrict__ hw, const float* __restrict__ hj,
                           const int* __restrict__ src, const int* __restrict__ dst,
                           float* __restrict__ out, int E) {
    int e = blockIdx.x * blockDim.x + threadIdx.x;
    if (e >= E) return;
    const float4* a = (const float4*)(hw + (size_t)src[e] * 16);
    const float4* b = (const float4*)(hj + (size_t)dst[e] * 16);
    float acc = 0.f;
#pragma unroll
    for (int i = 0; i < 4; ++i) {
        float4 x = a[i], y = b[i];
        acc = fmaf(x.x, y.x, acc);
        acc = fmaf(x.y, y.y, acc);
        acc = fmaf(x.z, y.z, acc);
        acc = fmaf(x.w, y.w, acc);
    }
    out[e] = acc;
}

// ---------------- host launch ----------------

static inline unsigned hgnn_blocks(size_t n) { return (unsigned)((n + TPB - 1) / TPB); }

extern "C" void kernel_launch(void* const* d_in, const int* in_sizes, int n_in,
                              void* d_out, int out_size, void* d_ws, size_t ws_size,
                              hipStream_t stream) {
    const float* job_feat    = (const float*)d_in[0];
    const float* worker_feat = (const float*)d_in[1];
    const int*   pre_src     = (const int*)d_in[2];
    const int*   pre_dst     = (const int*)d_in[3];
    const int*   nxt_src     = (const int*)d_in[4];
    const int*   nxt_dst     = (const int*)d_in[5];
    const int*   proc_src    = (const int*)d_in[6];
    const int*   proc_dst    = (const int*)d_in[7];
    const float* W_emb_job    = (const float*)d_in[8];
    const float* b_emb_job    = (const float*)d_in[9];
    const float* W_emb_worker = (const float*)d_in[10];
    const float* b_emb_worker = (const float*)d_in[11];
    const float* W_pre  = (const float*)d_in[12];
    const float* b_pre  = (const float*)d_in[13];
    const float* W_nxt  = (const float*)d_in[14];
    const float* b_nxt  = (const float*)d_in[15];
    const float* W_self = (const float*)d_in[16];
    const float* W_neigh= (const float*)d_in[17];
    const float* b_sage = (const float*)d_in[18];

    const int NJ = in_sizes[0] / 7;
    const int NW = in_sizes[1] / 3;
    const int EP = in_sizes[2];
    const int EN = in_sizes[4];
    const int EC = in_sizes[6];

    // workspace carve-out (256B aligned)
    char* w = (char*)d_ws;
    auto carve = [&](size_t bytes) {
        float* p = (float*)w;
        w += (bytes + 255) & ~(size_t)255;
        return p;
    };
    float* hj       = carve((size_t)NJ * 16 * sizeof(float));
    float* hw       = carve((size_t)NW * 16 * sizeof(float));
    float* aggs     = carve((size_t)NJ * 16 * 3 * sizeof(float)); // contiguous for one zero pass
    float* agg_pre  = aggs;
    float* agg_nxt  = aggs + (size_t)NJ * 16;
    float* agg_sage = aggs + (size_t)NJ * 32;
    float* scales   = carve((size_t)NJ * 5 * sizeof(float));      // contiguous for one zero pass
    float* rs_out_pre = scales;
    float* rs_in_pre  = scales + (size_t)NJ;
    float* rs_out_nxt = scales + (size_t)NJ * 2;
    float* rs_in_nxt  = scales + (size_t)NJ * 3;
    float* inv_cnt    = scales + (size_t)NJ * 4;

    // 1) degrees (computed once; edge lists are loop-invariant)
    {
        size_t n4 = (size_t)NJ * 5 / 4;
        hgnn_zero_f4<<<hgnn_blocks(n4), TPB, 0, stream>>>((float4*)scales, (unsigned)n4);
        hgnn_deg_pair<<<hgnn_blocks(EP), TPB, 0, stream>>>(pre_src, pre_dst, rs_out_pre, rs_in_pre, EP);
        hgnn_deg_pair<<<hgnn_blocks(EN), TPB, 0, stream>>>(nxt_src, nxt_dst, rs_out_nxt, rs_in_nxt, EN);
        hgnn_deg_dst<<<hgnn_blocks(EC), TPB, 0, stream>>>(proc_dst, inv_cnt, EC);
        hgnn_finalize_scales<<<hgnn_blocks(NJ), TPB, 0, stream>>>(rs_out_pre, rs_in_pre,
                                                                  rs_out_nxt, rs_in_nxt, inv_cnt, NJ);
    }

    // 2) embeddings
    hgnn_embed<<<hgnn_blocks((size_t)NJ * 16), TPB, 0, stream>>>(job_feat, W_emb_job, b_emb_job, hj, NJ, 7);
    hgnn_embed<<<hgnn_blocks((size_t)NW * 16), TPB, 0, stream>>>(worker_feat, W_emb_worker, b_emb_worker, hw, NW, 3);

    // 3) two message-passing rounds
    const int ntiles = (NJ + 15) >> 4;
    const unsigned upd_blocks = hgnn_blocks((size_t)ntiles * 32);
    for (int it = 0; it < 2; ++it) {
        size_t n4 = (size_t)NJ * 48 / 4;
        hgnn_zero_f4<<<hgnn_blocks(n4), TPB, 0, stream>>>((float4*)aggs, (unsigned)n4);
        hgnn_scatter<<<hgnn_blocks((size_t)EP * 16), TPB, 0, stream>>>(hj, pre_src, pre_dst, rs_out_pre,
                                                                       agg_pre, (unsigned)((size_t)EP * 16));
        hgnn_scatter<<<hgnn_blocks((size_t)EN * 16), TPB, 0, stream>>>(hj, nxt_src, nxt_dst, rs_out_nxt,
                                                                       agg_nxt, (unsigned)((size_t)EN * 16));
        hgnn_scatter<<<hgnn_blocks((size_t)EC * 16), TPB, 0, stream>>>(hw, proc_src, proc_dst, (const float*)nullptr,
                                                                       agg_sage, (unsigned)((size_t)EC * 16));
        hgnn_update_wmma<<<upd_blocks, TPB, 0, stream>>>(hj, agg_pre, agg_nxt, agg_sage,
                                                         rs_in_pre, rs_in_nxt, inv_cnt,
                                                         W_pre, W_nxt, W_self, W_neigh,
                                                         b_pre, b_nxt, b_sage, NJ);
    }

    // 4) dot-product readout
    hgnn_score<<<hgnn_blocks(EC), TPB, 0, stream>>>(hw, hj, proc_src, proc_dst, (float*)d_out, EC);
}